// integrator_39496519254624
// MI455X (gfx1250) — compile-verified
//
#include <hip/hip_runtime.h>

typedef __attribute__((ext_vector_type(2))) float v2f;
typedef __attribute__((ext_vector_type(4))) float v4f;
typedef __attribute__((ext_vector_type(8))) float v8f;

#define D_DIM 64
#define COLS 16
#define LDS_PITCH 68   // 64 + 4 pad: bank-conflict-free col-major tile

// DS-only workgroup barrier (CDNA5 split barrier).
// Unlike __syncthreads(), this does NOT wait on STOREcnt/LOADcnt, so the
// per-step global trajectory stores overlap with subsequent compute.
// s_wait_dscnt 0 before the signal guarantees this wave's LDS writes are
// visible and its LDS reads have completed before any wave passes the wait.
__device__ __forceinline__ void lds_barrier()
{
    asm volatile("s_wait_dscnt 0x0\n\t"
                 "s_barrier_signal -1\n\t"
                 "s_barrier_wait -1" ::: "memory");
}

// One evaluation of f(x) = x * (r + A @ x) for this wave's 16x16 tile.
// xs   : stage input, C-layout (lane holds rows rowbase..rowbase+7, col ln)
// af   : A fragments (16x4 tiles, K-blocks of 4), ISA A layout
// rf   : growth rates in C layout
// buf  : LDS staging tile for THIS stage (caller double-buffers: only one
//        barrier per stage is needed; the next stage's barrier protects
//        this buffer from being overwritten before all reads completed)
__device__ __forceinline__ void glv_f(const float xs[8], const v2f af[16],
                                      const float rf[8], float kout[8],
                                      float* buf, int ln, int hf, int rowbase)
{
    // publish stage input to LDS (column-major, padded pitch)
    float* wp = buf + ln * LDS_PITCH + rowbase;
    *(v4f*)(wp)     = (v4f){xs[0], xs[1], xs[2], xs[3]};
    *(v4f*)(wp + 4) = (v4f){xs[4], xs[5], xs[6], xs[7]};
    lds_barrier();

    // gather B fragments (4x16 tiles: lane ln, component v -> row 4k+v+2*hf)
    const float* rp = buf + ln * LDS_PITCH + 2 * hf;
    v8f c0 = {0.f, 0.f, 0.f, 0.f, 0.f, 0.f, 0.f, 0.f};
    v8f c1 = {0.f, 0.f, 0.f, 0.f, 0.f, 0.f, 0.f, 0.f};
#pragma unroll
    for (int k = 0; k < 16; k += 2) {
        v2f b0 = *(const v2f*)(rp + 4 * k);
        v2f b1 = *(const v2f*)(rp + 4 * (k + 1));
        c0 = __builtin_amdgcn_wmma_f32_16x16x4_f32(false, af[k],     false, b0,
                                                   (short)0, c0, false, false);
        c1 = __builtin_amdgcn_wmma_f32_16x16x4_f32(false, af[k + 1], false, b1,
                                                   (short)0, c1, false, false);
    }

#pragma unroll
    for (int j = 0; j < 8; ++j)
        kout[j] = xs[j] * (rf[j] + (c0[j] + c1[j]));
}

__launch_bounds__(128)
__global__ void glv_rk4_wmma(const float* __restrict__ x0,
                             const float* __restrict__ r,
                             const float* __restrict__ A,
                             const float* __restrict__ tgrid,
                             float* __restrict__ out,
                             int nsteps)
{
    __shared__ float xS[2][COLS * LDS_PITCH];   // double-buffered stage tile

    const int tid  = threadIdx.x;
    const int wave = tid >> 5;        // M-tile 0..3
    const int lane = tid & 31;
    const int hf   = lane >> 4;       // lane half
    const int ln   = lane & 15;       // column within tile / m within tile
    const int col0 = blockIdx.x * COLS;
    const int rowbase = wave * 16 + 8 * hf;   // first of 8 consecutive rows

    const float dt  = tgrid[1] - tgrid[0];
    const float hdt = 0.5f * dt;
    const float sdt = dt * (1.0f / 6.0f);

    // --- A fragments: a[k] = { A[m][4k+2*hf], A[m][4k+2*hf+1] }, m = wave*16+ln
    v2f af[16];
    {
        const float* arow = A + (wave * 16 + ln) * D_DIM + 2 * hf;
#pragma unroll
        for (int k = 0; k < 16; ++k) {
            af[k].x = arow[4 * k + 0];
            af[k].y = arow[4 * k + 1];
        }
    }

    // --- r fragment in C layout
    float rf[8];
#pragma unroll
    for (int j = 0; j < 8; ++j) rf[j] = r[rowbase + j];

    // --- initial state tile, C layout: lane holds x0[col0+ln][rowbase..rowbase+7]
    float x[8];
    {
        const float* xp = x0 + (size_t)(col0 + ln) * D_DIM + rowbase;
#pragma unroll
        for (int j = 0; j < 8; ++j) x[j] = xp[j];
    }

    // --- t = 0 trajectory row
    float* outp = out + ((size_t)(col0 + ln) * nsteps) * D_DIM + rowbase;
    *(v4f*)(outp)     = (v4f){x[0], x[1], x[2], x[3]};
    *(v4f*)(outp + 4) = (v4f){x[4], x[5], x[6], x[7]};

    float k1[8], k2[8], k3[8], k4[8], xs[8];
    for (int t = 1; t < nsteps; ++t) {
        // 4 stages alternate LDS buffers: a, b, a, b (4 barriers per step)
        glv_f(x, af, rf, k1, xS[0], ln, hf, rowbase);
#pragma unroll
        for (int j = 0; j < 8; ++j) xs[j] = x[j] + hdt * k1[j];
        glv_f(xs, af, rf, k2, xS[1], ln, hf, rowbase);
#pragma unroll
        for (int j = 0; j < 8; ++j) xs[j] = x[j] + hdt * k2[j];
        glv_f(xs, af, rf, k3, xS[0], ln, hf, rowbase);
#pragma unroll
        for (int j = 0; j < 8; ++j) xs[j] = x[j] + dt * k3[j];
        glv_f(xs, af, rf, k4, xS[1], ln, hf, rowbase);
#pragma unroll
        for (int j = 0; j < 8; ++j)
            x[j] += sdt * (k1[j] + 2.0f * (k2[j] + k3[j]) + k4[j]);

        float* op = outp + (size_t)t * D_DIM;
        *(v4f*)(op)     = (v4f){x[0], x[1], x[2], x[3]};
        *(v4f*)(op + 4) = (v4f){x[4], x[5], x[6], x[7]};
    }
}

extern "C" void kernel_launch(void* const* d_in, const int* in_sizes, int n_in,
                              void* d_out, int out_size, void* d_ws, size_t ws_size,
                              hipStream_t stream)
{
    const float* x0 = (const float*)d_in[0];   // (BATCH, 64)
    const float* r  = (const float*)d_in[1];   // (64,)
    const float* A  = (const float*)d_in[2];   // (64, 64)
    const float* tg = (const float*)d_in[3];   // (N_TSTEPS,)

    float* out = (float*)d_out;                // (BATCH, N_TSTEPS, 64)

    const int batch  = in_sizes[0] / D_DIM;
    const int nsteps = in_sizes[3];

    dim3 grid(batch / COLS);
    dim3 block(128);
    hipLaunchKernelGGL(glv_rk4_wmma, grid, block, 0, stream,
                       x0, r, A, tg, out, nsteps);
}